// ModulatedDeformConvPack_71287867179021
// MI455X (gfx1250) — compile-verified
//
#include <hip/hip_runtime.h>
#include <math.h>

typedef __attribute__((ext_vector_type(2))) float v2f;
typedef __attribute__((ext_vector_type(8))) float v8f;

namespace dcn {
constexpr int BN    = 2;     // batch
constexpr int CI    = 64;    // input channels
constexpr int CO    = 64;    // output channels
constexpr int HH    = 128;
constexpr int WW    = 128;
constexpr int DG    = 8;     // deform groups
constexpr int CPG   = CI / DG;      // 8 channels per group
constexpr int KT    = 9;     // 3x3 taps
constexpr int OFFC  = DG * 3 * KT;  // 216 real offset-conv channels
constexpr int OFFCP = 224;          // padded to 14 full 16-row M-tiles
constexpr int KDIM  = CI * KT;      // 576 GEMM K
constexpr int LDSS  = KDIM + 8;     // padded LDS col stride (584, 8B-aligned pairs)
constexpr int TPB   = 128;          // 4 waves of 32
}

__device__ __forceinline__ float sigmoid_f(float z) {
    return 1.0f / (1.0f + __expf(-z));
}

__device__ __forceinline__ float samp_f(const float* __restrict__ xc, int y, int x) {
    using namespace dcn;
    if (y < 0 || y >= HH || x < 0 || x >= WW) return 0.0f;
    return xc[y * WW + x];
}

// ---------------------------------------------------------------------------
// Kernel 1: offset conv  scratch(B,224,H,W) = conv3x3(x, conv_offset_w) + b
// Implicit GEMM, M = 216(+8 pad) offset channels, N = 16 pixels, K = 576.
// Channels 216..223 are padding: computed (clamped weights) but never read.
// ---------------------------------------------------------------------------
__global__ __launch_bounds__(dcn::TPB) void dcn_offset_conv_kernel(
    const float* __restrict__ x,
    const float* __restrict__ wof,    // (216, 64, 3, 3) == (216, 576) row-major
    const float* __restrict__ bof,    // (216,)
    float* __restrict__ offBuf)       // (B, 224, H, W) scratch
{
    using namespace dcn;
    __shared__ float col[16 * LDSS];

    const int tile  = blockIdx.x;                 // over (b, h, w/16)
    const int b     = tile / (HH * (WW / 16));
    const int rem   = tile % (HH * (WW / 16));
    const int h     = rem / (WW / 16);
    const int wBase = (rem % (WW / 16)) * 16;

    const int t   = threadIdx.x;
    const int pix = t >> 3;        // 0..15
    const int sub = t & 7;         // 0..7
    const int w   = wBase + pix;

    // ---- phase 1: zero-padded im2col tile col[pix][cin*9 + k] ----
    const float* xb = x + (size_t)b * CI * HH * WW;
    for (int j = 0; j < CI / 8; ++j) {
        const int cin = j * 8 + sub;
        const float* xc = xb + (size_t)cin * HH * WW;
        float* cp = &col[pix * LDSS + cin * KT];
#pragma unroll
        for (int kh = 0; kh < 3; ++kh) {
            const int y = h - 1 + kh;
            const bool yok = (y >= 0) && (y < HH);
#pragma unroll
            for (int kw = 0; kw < 3; ++kw) {
                const int xx = w - 1 + kw;
                const bool ok = yok && (xx >= 0) && (xx < WW);
                cp[kh * 3 + kw] = ok ? xc[y * WW + xx] : 0.0f;
            }
        }
    }
    __syncthreads();

    // ---- phase 2: WMMA f32 16x16x4, 14 M-tiles over 4 waves ----
    const int wv   = t >> 5;
    const int lane = t & 31;
    const int row  = lane & 15;    // M row for A, N column (pixel) for B
    const int hi   = lane >> 4;
    const int koff = hi * 2;
    const float* colRow = &col[row * LDSS];

    for (int mt = wv; mt < OFFCP / 16; mt += 4) {
        const int ocBase = mt * 16;
        const int ocr    = ocBase + row;
        const int occ    = ocr < OFFC ? ocr : OFFC - 1;   // clamp OOB A loads
        const float* aRow = wof + (size_t)occ * KDIM;
        __builtin_prefetch(aRow, 0, 1);

        v8f acc = {};
#pragma unroll 4
        for (int k0 = 0; k0 < KDIM; k0 += 4) {
            v2f a  = *(const v2f*)(aRow + k0 + koff);
            v2f bb = *(const v2f*)(colRow + k0 + koff);
            acc = __builtin_amdgcn_wmma_f32_16x16x4_f32(
                false, a, false, bb, (short)0, acc, false, false);
        }

        const int wn = wBase + row;   // lane's pixel column
#pragma unroll
        for (int r = 0; r < 8; ++r) {
            const int o  = ocBase + r + hi * 8;           // < 224, unconditional
            const int ob = o < OFFC ? o : OFFC - 1;       // clamp bias read only
            offBuf[(((size_t)b * OFFCP + o) * HH + h) * WW + wn] = acc[r] + bof[ob];
        }
    }
}

// ---------------------------------------------------------------------------
// Kernel 2: deformable sampling + main GEMM
// col K-index = cin*9 + tap matches weight (64, 64, 3, 3) == (64, 576) layout.
// ---------------------------------------------------------------------------
__global__ __launch_bounds__(dcn::TPB) void dcn_main_kernel(
    const float* __restrict__ x,
    const float* __restrict__ offBuf,   // (B, 224, H, W) raw conv + bias
    const float* __restrict__ wmain,    // (64, 576)
    const float* __restrict__ bias,     // (64,)
    float* __restrict__ out)            // (B, 64, H, W)
{
    using namespace dcn;
    __shared__ float col[16 * LDSS];

    const int tile  = blockIdx.x;
    const int b     = tile / (HH * (WW / 16));
    const int rem   = tile % (HH * (WW / 16));
    const int h     = rem / (WW / 16);
    const int wBase = (rem % (WW / 16)) * 16;

    const int t   = threadIdx.x;
    const int pix = t >> 3;
    const int sub = t & 7;         // channel-in-group
    const int w   = wBase + pix;

    const float* xb = x + (size_t)b * CI * HH * WW;
    const float* ob = offBuf + (size_t)b * OFFCP * HH * WW;
    const int hw = h * WW + w;

    // ---- phase 1: bilinear-sampled, mask-modulated col tile ----
    for (int gk = 0; gk < DG * KT; ++gk) {     // 72 (group, tap) pairs
        const int g  = gk / KT;
        const int k  = gk % KT;
        const int kh = k / 3;
        const int kw = k % 3;

        const float offy = ob[(size_t)(g * KT + k) * HH * WW + hw];
        const float offx = ob[(size_t)(DG * KT + g * KT + k) * HH * WW + hw];
        const float m    = sigmoid_f(ob[(size_t)(2 * DG * KT + g * KT + k) * HH * WW + hw]);

        const float py = offy + (float)(h - 1 + kh);
        const float px = offx + (float)(w - 1 + kw);
        const float y0f = floorf(py);
        const float x0f = floorf(px);
        const float ly = py - y0f;
        const float lx = px - x0f;
        const int y0 = (int)y0f;
        const int x0 = (int)x0f;

        const int cin = g * CPG + sub;
        const float* xc = xb + (size_t)cin * HH * WW;
        const float v00 = samp_f(xc, y0,     x0);
        const float v01 = samp_f(xc, y0,     x0 + 1);
        const float v10 = samp_f(xc, y0 + 1, x0);
        const float v11 = samp_f(xc, y0 + 1, x0 + 1);
        const float hy = 1.0f - ly, hx = 1.0f - lx;
        const float val = hy * hx * v00 + hy * lx * v01 + ly * hx * v10 + ly * lx * v11;

        col[pix * LDSS + cin * KT + k] = val * m;
    }
    __syncthreads();

    // ---- phase 2: one 16x16 M-tile per wave (64 output channels / 4 waves) ----
    const int wv   = t >> 5;
    const int lane = t & 31;
    const int row  = lane & 15;
    const int hi   = lane >> 4;
    const int koff = hi * 2;
    const int ocBase = wv * 16;
    const float* aRow   = wmain + (size_t)(ocBase + row) * KDIM;
    const float* colRow = &col[row * LDSS];
    __builtin_prefetch(aRow, 0, 1);

    v8f acc = {};
#pragma unroll 4
    for (int k0 = 0; k0 < KDIM; k0 += 4) {
        v2f a  = *(const v2f*)(aRow + k0 + koff);
        v2f bb = *(const v2f*)(colRow + k0 + koff);
        acc = __builtin_amdgcn_wmma_f32_16x16x4_f32(
            false, a, false, bb, (short)0, acc, false, false);
    }

    const int wn = wBase + row;
#pragma unroll
    for (int r = 0; r < 8; ++r) {
        const int o = ocBase + r + hi * 8;
        out[(((size_t)b * CO + o) * HH + h) * WW + wn] = acc[r] + bias[o];
    }
}

// ---------------------------------------------------------------------------
extern "C" void kernel_launch(void* const* d_in, const int* in_sizes, int n_in,
                              void* d_out, int out_size, void* d_ws, size_t ws_size,
                              hipStream_t stream) {
    using namespace dcn;
    const float* x     = (const float*)d_in[0];
    const float* wof   = (const float*)d_in[1];
    const float* bof   = (const float*)d_in[2];
    const float* wmain = (const float*)d_in[3];
    const float* bias  = (const float*)d_in[4];
    float* out    = (float*)d_out;
    float* offBuf = (float*)d_ws;   // B*224*H*W fp32 = ~29.4 MB scratch

    const dim3 grid(BN * HH * (WW / 16));   // 2048 tiles of 16 pixels
    const dim3 block(TPB);

    hipLaunchKernelGGL(dcn_offset_conv_kernel, grid, block, 0, stream,
                       x, wof, bof, offBuf);
    hipLaunchKernelGGL(dcn_main_kernel, grid, block, 0, stream,
                       x, offBuf, wmain, bias, out);
}